// MultiplePolyConvFrame_7224134992559
// MI455X (gfx1250) — compile-verified
//
#include <hip/hip_runtime.h>

#define N_NODE 50000
#define N_EDGE 800000
#define D_FEAT 64
#define M_CH   11

typedef __attribute__((ext_vector_type(2))) float v2f;
typedef __attribute__((ext_vector_type(8))) float v8f;

// ---------------- CSR build ----------------

__global__ void count_rows(const int* __restrict__ row, int* __restrict__ counts) {
  int e = blockIdx.x * blockDim.x + threadIdx.x;
  if (e < N_EDGE) atomicAdd(&counts[row[e]], 1);
}

__global__ void scan_blocks(const int* __restrict__ counts, int* __restrict__ offs,
                            int* __restrict__ bsums) {
  __shared__ int sm[256];
  int tid = threadIdx.x;
  int gid = blockIdx.x * 256 + tid;
  int v = (gid < N_NODE) ? counts[gid] : 0;
  sm[tid] = v;
  __syncthreads();
  for (int o = 1; o < 256; o <<= 1) {
    int t = (tid >= o) ? sm[tid - o] : 0;
    __syncthreads();
    sm[tid] += t;
    __syncthreads();
  }
  if (gid < N_NODE) offs[gid] = sm[tid] - v;   // exclusive within block
  if (tid == 255) bsums[blockIdx.x] = sm[255];
}

__global__ void scan_sums(int* bsums) {
  __shared__ int sm[256];
  int tid = threadIdx.x;
  int v = bsums[tid];
  sm[tid] = v;
  __syncthreads();
  for (int o = 1; o < 256; o <<= 1) {
    int t = (tid >= o) ? sm[tid - o] : 0;
    __syncthreads();
    sm[tid] += t;
    __syncthreads();
  }
  bsums[tid] = sm[tid] - v;                    // exclusive
}

__global__ void finalize_offs(const int* __restrict__ counts, const int* __restrict__ bsums,
                              int* __restrict__ offs, float* __restrict__ dinv) {
  int gid = blockIdx.x * 256 + threadIdx.x;    // must match scan_blocks partition
  if (gid < N_NODE) {
    offs[gid] += bsums[blockIdx.x];
    float dg = (float)counts[gid];
    if (dg < 0.5f) dg += 1.0f;
    dinv[gid] = 1.0f / sqrtf(dg);
  }
  if (gid == 0) offs[N_NODE] = N_EDGE;
}

__global__ void scatter_edges(const int* __restrict__ row, const int* __restrict__ col,
                              const float* __restrict__ attr, const float* __restrict__ dinv,
                              const int* __restrict__ offs, int* __restrict__ cursor,
                              int* __restrict__ cs, float* __restrict__ vs) {
  int e = blockIdx.x * blockDim.x + threadIdx.x;
  if (e >= N_EDGE) return;
  int r = row[e], c = col[e];
  int p = offs[r] + atomicAdd(&cursor[r], 1);
  cs[p] = c;
  vs[p] = dinv[r] * attr[e] * dinv[c];
}

// ---------------- recurrence coefficients ----------------
// cf layout: cf[((L-1)*M + m)*3 + {0,1,2}] = {cA (spmm term), cB (prev), cC (prevprev)}

__global__ void coefs_kernel(const float* __restrict__ alphas, const float* __restrict__ a_arr,
                             const float* __restrict__ b_arr, float* __restrict__ cf) {
  int m = threadIdx.x;
  if (m >= M_CH) return;
  const float l = -1.0f, r = 1.0f;
  float a = a_arr[m], b = b_arr[m];
  float al0 = alphas[0 * M_CH + m];
  float coef1 = (a - b) * 0.5f - (a + b + 2.0f) * 0.5f * (l + r) / (r - l);
  float coef2 = (a + b + 2.0f) / (r - l);
  cf[(0 * M_CH + m) * 3 + 0] = al0 * coef2;
  cf[(0 * M_CH + m) * 3 + 1] = al0 * coef1;
  cf[(0 * M_CH + m) * 3 + 2] = 0.0f;
  for (int L = 2; L <= 3; ++L) {
    float Lf = (float)L;
    float coef_l   = 2.0f * Lf * (Lf + a + b) * (2.0f * Lf - 2.0f + a + b);
    float c1 = (2.0f * Lf + a + b - 1.0f) * (2.0f * Lf + a + b) * (2.0f * Lf + a + b - 2.0f);
    float c2 = (2.0f * Lf + a + b - 1.0f) * (a * a - b * b);
    float c3 = 2.0f * (Lf - 1.0f + a) * (Lf - 1.0f + b) * (2.0f * Lf + a + b);
    float alL   = alphas[(L - 1) * M_CH + m];
    float alLm1 = alphas[(L - 2) * M_CH + m];
    float tmp1 = alL * (c1 / coef_l);
    float tmp2 = alL * (c2 / coef_l);
    float tmp3 = alL * alLm1 * (c3 / coef_l);
    float t12 = tmp1 * (2.0f / (r - l));
    float t22 = tmp1 * ((r + l) / (r - l)) + tmp2;
    cf[((L - 1) * M_CH + m) * 3 + 0] = t12;
    cf[((L - 1) * M_CH + m) * 3 + 1] = -t22;
    cf[((L - 1) * M_CH + m) * 3 + 2] = -tmp3;
  }
}

// ---------------- SpMM core: one wave32 per node, 2 feats per lane ----------------

__device__ inline float2 spmm_row(const float* __restrict__ xsrc, const int* __restrict__ cs,
                                  const float* __restrict__ vs, int start, int end, int lane) {
  float2 acc = make_float2(0.0f, 0.0f);
  for (int j0 = start; j0 < end; j0 += 32) {
    int jj = j0 + lane;
    int c = 0; float v = 0.0f;
    if (jj < end) { c = cs[jj]; v = vs[jj]; }        // coalesced edge loads
    int cnt = end - j0; if (cnt > 32) cnt = 32;
    for (int t = 0; t < cnt; ++t) {
      int   cc = __shfl(c, t, 32);                   // broadcast within wave32
      float vv = __shfl(v, t, 32);
      float2 xv = *((const float2*)(xsrc + cc * D_FEAT) + lane);  // L2-resident gather
      acc.x = fmaf(vv, xv.x, acc.x);
      acc.y = fmaf(vv, xv.y, acc.y);
    }
  }
  return acc;
}

__global__ void spmm_x(const float* __restrict__ x, const int* __restrict__ offs,
                       const int* __restrict__ cs, const float* __restrict__ vs,
                       float* __restrict__ s) {
  int wave = (blockIdx.x * blockDim.x + threadIdx.x) >> 5;
  int lane = threadIdx.x & 31;
  if (wave >= N_NODE) return;
  float2 acc = spmm_row(x, cs, vs, offs[wave], offs[wave + 1], lane);
  *((float2*)(s + (size_t)wave * D_FEAT) + lane) = acc;
}

// level-1 combine: xs1[m] = cA*s + cB*x  (s shared across channels)
__global__ void lvl1_combine(const float* __restrict__ x, const float* __restrict__ s,
                             const float* __restrict__ cf, float* __restrict__ cur) {
  int idx = blockIdx.x * blockDim.x + threadIdx.x;   // over M*N*(64/4) float4s
  const int per = N_NODE * (D_FEAT / 4);
  int m = idx / per;
  int i = idx - m * per;
  float cA = cf[m * 3 + 0], cB = cf[m * 3 + 1];
  float4 xv = ((const float4*)x)[i];
  float4 sv = ((const float4*)s)[i];
  float4 o;
  o.x = cA * sv.x + cB * xv.x;
  o.y = cA * sv.y + cB * xv.y;
  o.z = cA * sv.z + cB * xv.z;
  o.w = cA * sv.w + cB * xv.w;
  ((float4*)cur)[(size_t)m * per + i] = o;
}

// level>=2: cur[m] = cA*spmm(prev[m]) + cB*prev[m] + cC*prevprev[m]
// (no __restrict__ on pp/cur: level 3 ping-pongs pp==cur, each location touched
//  only by its owning lane, read-before-write within the thread)
__global__ void spmm_combine(const float* __restrict__ prev, const float* pp, long ppStride,
                             const float* __restrict__ cf, float* cur,
                             const int* __restrict__ offs, const int* __restrict__ cs,
                             const float* __restrict__ vs) {
  int wave = (blockIdx.x * blockDim.x + threadIdx.x) >> 5;
  int lane = threadIdx.x & 31;
  if (wave >= M_CH * N_NODE) return;
  int m = wave / N_NODE;
  int n = wave - m * N_NODE;
  const float* pm = prev + (size_t)m * N_NODE * D_FEAT;
  float2 acc = spmm_row(pm, cs, vs, offs[n], offs[n + 1], lane);
  float cA = cf[m * 3 + 0], cB = cf[m * 3 + 1], cC = cf[m * 3 + 2];
  float2 p = *((const float2*)(pm + (size_t)n * D_FEAT) + lane);
  float2 q = *((const float2*)(pp + (size_t)m * ppStride + (size_t)n * D_FEAT) + lane);
  float2 o;
  o.x = cA * acc.x + cB * p.x + cC * q.x;
  o.y = cA * acc.y + cB * p.y + cC * q.y;
  *((float2*)(cur + (size_t)m * N_NODE * D_FEAT + (size_t)n * D_FEAT) + lane) = o;
}

// ---------------- channel reduction via f32 WMMA ----------------
// out[n,lvl,d] = sum_m w[m]*buf[m][n][d].  D = A x B + C with
// A[i,k] = buf[m=k][ (n,d-tile) row i ]  (16x4 f32, doc layout: V0=K0/K2, V1=K1/K3)
// B[k,j] = w[m=k] broadcast over columns j. Column 0 of D = desired 16 sums
// (lanes 0 and 16 hold rows 0-7 / 8-15 as v8f).

__global__ void sum_channels_wmma(const float* __restrict__ buf, const float* __restrict__ w,
                                  float* __restrict__ out, int lvl) {
  int wave = (blockIdx.x * blockDim.x + threadIdx.x) >> 5;
  int lane = threadIdx.x & 31;
  if (wave >= N_NODE * 4) return;            // 4 d-tiles of 16 per node
  int n  = wave >> 2;
  int dt = wave & 3;
  int row = lane & 15;
  int klo = (lane < 16) ? 0 : 2;
  size_t src = (size_t)n * D_FEAT + dt * 16 + row;
  const size_t chStride = (size_t)N_NODE * D_FEAT;
  v8f c = {0.f, 0.f, 0.f, 0.f, 0.f, 0.f, 0.f, 0.f};
#pragma unroll
  for (int mb = 0; mb < 12; mb += 4) {       // M=11 padded to 12
    int m0 = mb + klo, m1 = m0 + 1;
    v2f a, b;
    a.x = (m0 < M_CH) ? buf[(size_t)m0 * chStride + src] : 0.0f;
    a.y = (m1 < M_CH) ? buf[(size_t)m1 * chStride + src] : 0.0f;
    b.x = (m0 < M_CH) ? w[m0] : 0.0f;
    b.y = (m1 < M_CH) ? w[m1] : 0.0f;
    c = __builtin_amdgcn_wmma_f32_16x16x4_f32(false, a, false, b, (short)0, c, false, false);
  }
  if ((lane & 15) == 0) {                    // lanes 0 and 16 hold column 0
    int rbase = (lane >> 4) * 8;
    float* o = out + ((size_t)n * 4 + lvl) * D_FEAT + dt * 16 + rbase;
    *(v8f*)o = c;                            // 32B aligned store
  }
}

// level 0: out[n,0,d] = (sum_m w[m]) * x[n,d]
__global__ void lvl0_out(const float* __restrict__ x, const float* __restrict__ w,
                         float* __restrict__ out) {
  int idx = blockIdx.x * blockDim.x + threadIdx.x;   // over N*(64/4) float4s
  if (idx >= N_NODE * (D_FEAT / 4)) return;
  float ws = 0.0f;
  for (int m = 0; m < M_CH; ++m) ws += w[m];
  int n = idx >> 4;
  int dq = idx & 15;
  float4 xv = ((const float4*)x)[idx];
  float4 o;
  o.x = ws * xv.x; o.y = ws * xv.y; o.z = ws * xv.z; o.w = ws * xv.w;
  ((float4*)(out + (size_t)n * 4 * D_FEAT))[dq] = o;
}

// ---------------- host-side launch ----------------

extern "C" void kernel_launch(void* const* d_in, const int* in_sizes, int n_in,
                              void* d_out, int out_size, void* d_ws, size_t ws_size,
                              hipStream_t stream) {
  const float* x      = (const float*)d_in[0];
  const int*   ei     = (const int*)  d_in[1];
  const float* attr   = (const float*)d_in[2];
  const float* alphas = (const float*)d_in[3];
  const float* w      = (const float*)d_in[4];
  const float* a_arr  = (const float*)d_in[5];
  const float* b_arr  = (const float*)d_in[6];
  float* out = (float*)d_out;
  const int* row = ei;
  const int* col = ei + N_EDGE;

  char* base = (char*)d_ws;
  size_t off = 0;
  auto alloc = [&](size_t bytes) -> void* {
    void* p = base + off;
    off = (off + bytes + 63) & ~(size_t)63;
    return p;
  };
  int*   counts = (int*)  alloc((size_t)N_NODE * 4);
  int*   cursor = (int*)  alloc((size_t)N_NODE * 4);
  int*   bsums  = (int*)  alloc(256 * 4);
  int*   offs   = (int*)  alloc((size_t)(N_NODE + 1) * 4);
  float* dinv   = (float*)alloc((size_t)N_NODE * 4);
  int*   cs     = (int*)  alloc((size_t)N_EDGE * 4);
  float* vs     = (float*)alloc((size_t)N_EDGE * 4);
  float* cf     = (float*)alloc(3 * M_CH * 3 * 4);
  float* s      = (float*)alloc((size_t)N_NODE * D_FEAT * 4);
  float* bufA   = (float*)alloc((size_t)M_CH * N_NODE * D_FEAT * 4);
  float* bufB   = (float*)alloc((size_t)M_CH * N_NODE * D_FEAT * 4);

  hipMemsetAsync(counts, 0, (size_t)N_NODE * 4, stream);
  hipMemsetAsync(cursor, 0, (size_t)N_NODE * 4, stream);
  hipMemsetAsync(bsums,  0, 256 * 4, stream);

  count_rows   <<<(N_EDGE + 255) / 256, 256, 0, stream>>>(row, counts);
  scan_blocks  <<<196, 256, 0, stream>>>(counts, offs, bsums);
  scan_sums    <<<1,   256, 0, stream>>>(bsums);
  finalize_offs<<<196, 256, 0, stream>>>(counts, bsums, offs, dinv);
  scatter_edges<<<(N_EDGE + 255) / 256, 256, 0, stream>>>(row, col, attr, dinv, offs, cursor, cs, vs);
  coefs_kernel <<<1, 32, 0, stream>>>(alphas, a_arr, b_arr, cf);

  // level 1 (spmm(x) shared by all channels)
  spmm_x       <<<N_NODE / 8, 256, 0, stream>>>(x, offs, cs, vs, s);
  lvl1_combine <<<(M_CH * N_NODE * (D_FEAT / 4)) / 256, 256, 0, stream>>>(x, s, cf, bufA);
  sum_channels_wmma<<<(N_NODE * 4) / 8, 256, 0, stream>>>(bufA, w, out, 1);

  // level 2 (prevprev = x, channel stride 0)
  spmm_combine <<<(M_CH * N_NODE) / 8, 256, 0, stream>>>(bufA, x, 0L, cf + M_CH * 3, bufB,
                                                         offs, cs, vs);
  sum_channels_wmma<<<(N_NODE * 4) / 8, 256, 0, stream>>>(bufB, w, out, 2);

  // level 3 (ping-pong: overwrite bufA)
  spmm_combine <<<(M_CH * N_NODE) / 8, 256, 0, stream>>>(bufB, bufA, (long)N_NODE * D_FEAT,
                                                         cf + 2 * M_CH * 3, bufA, offs, cs, vs);
  sum_channels_wmma<<<(N_NODE * 4) / 8, 256, 0, stream>>>(bufA, w, out, 3);

  // level 0
  lvl0_out     <<<(N_NODE * (D_FEAT / 4)) / 256, 256, 0, stream>>>(x, w, out);
}